// WordGAT_1082331759097
// MI455X (gfx1250) — compile-verified
//
#include <hip/hip_runtime.h>
#include <math.h>

#define NW 100000
#define NS 10000
#define NE 800000
#define DD 256
#define HH 4
#define DHD 64
#define DF 512
#define NEG_SLOPE 0.2f

typedef float v2f __attribute__((ext_vector_type(2)));
typedef float v8f __attribute__((ext_vector_type(8)));

// ---------------------------------------------------------------------------
// WMMA fp32 GEMM: C[M,N] = A[M,K] @ B[K,N] (+row-bias, optional ReLU).
// One wave computes a 32x32 output tile: 2 A-fragments x 2 B-fragments ->
// 4 independent accumulators, 4 v_wmma_f32_16x16x4_f32 per K-step.
// Requires N%32==0, K%4==0. M handled with row clamps (WMMA needs EXEC=all-1s,
// so we never predicate lanes; we clamp A row loads and guard stores).
// ---------------------------------------------------------------------------
__global__ __launch_bounds__(32)
void gemm_wmma_f32(const float* __restrict__ A, const float* __restrict__ B,
                   const float* __restrict__ bias, float* __restrict__ C,
                   int M, int N, int K, int fuse_relu)
{
    const int lane = threadIdx.x & 31;
    const int half = lane >> 4;     // 0: lanes 0-15, 1: lanes 16-31
    const int lrow = lane & 15;
    const int tn = blockIdx.x * 32;
    const int tm = blockIdx.y * 32;

    // A rows for the two M-halves of the tile (clamped: harmless dup loads)
    const int ar0 = min(tm + lrow,      M - 1);
    const int ar1 = min(tm + 16 + lrow, M - 1);
    const float* __restrict__ Arow0 = A + (size_t)ar0 * K;
    const float* __restrict__ Arow1 = A + (size_t)ar1 * K;
    // B columns for the two N-halves
    const float* __restrict__ Bc0 = B + (tn + lrow);
    const float* __restrict__ Bc1 = B + (tn + 16 + lrow);

    v8f acc00 = {}, acc01 = {}, acc10 = {}, acc11 = {};
#pragma unroll 2
    for (int k = 0; k < K; k += 4) {
        const int kk = k + 2 * half;
        // A 16x4 fragments: lane<16 holds K=k,k+1 ; lane>=16 holds K=k+2,k+3
        v2f a0 = *(const v2f*)(Arow0 + kk);
        v2f a1 = *(const v2f*)(Arow1 + kk);
        // B 4x16 fragments: VGPR0 = row kk, VGPR1 = row kk+1
        const size_t r0 = (size_t)kk * N;
        const size_t r1 = r0 + N;
        v2f b0; b0.x = Bc0[r0]; b0.y = Bc0[r1];
        v2f b1; b1.x = Bc1[r0]; b1.y = Bc1[r1];
        acc00 = __builtin_amdgcn_wmma_f32_16x16x4_f32(false, a0, false, b0, (short)0, acc00, false, false);
        acc01 = __builtin_amdgcn_wmma_f32_16x16x4_f32(false, a0, false, b1, (short)0, acc01, false, false);
        acc10 = __builtin_amdgcn_wmma_f32_16x16x4_f32(false, a1, false, b0, (short)0, acc10, false, false);
        acc11 = __builtin_amdgcn_wmma_f32_16x16x4_f32(false, a1, false, b1, (short)0, acc11, false, false);
    }

    // C/D layout: VGPR j -> M = tile_m + j + 8*half, N = tile_n + lrow
    const int col0 = tn + lrow;
    const int col1 = tn + 16 + lrow;
    const float bi0 = bias ? bias[col0] : 0.f;
    const float bi1 = bias ? bias[col1] : 0.f;
#pragma unroll
    for (int j = 0; j < 8; ++j) {
        const int r0 = tm + j + 8 * half;      // upper 16-row half
        const int r1 = r0 + 16;                // lower 16-row half
        if (r0 < M) {
            float v = acc00[j] + bi0;
            if (fuse_relu) v = fmaxf(v, 0.f);
            C[(size_t)r0 * N + col0] = v;
            v = acc01[j] + bi1;
            if (fuse_relu) v = fmaxf(v, 0.f);
            C[(size_t)r0 * N + col1] = v;
        }
        if (r1 < M) {
            float v = acc10[j] + bi0;
            if (fuse_relu) v = fmaxf(v, 0.f);
            C[(size_t)r1 * N + col0] = v;
            v = acc11[j] + bi1;
            if (fuse_relu) v = fmaxf(v, 0.f);
            C[(size_t)r1 * N + col1] = v;
        }
    }
}

// ---------------------------------------------------------------------------
// el[n,h] = sum_j feat_src[n, h*64+j] * attn_l[h,j]
// ---------------------------------------------------------------------------
__global__ void el_kernel(const float* __restrict__ feat, const float* __restrict__ attn,
                          float* __restrict__ out, int n_total)
{
    int i = blockIdx.x * blockDim.x + threadIdx.x;   // i = node*H + h
    if (i >= n_total) return;
    int h = i & (HH - 1);
    int node = i >> 2;
    const float* f = feat + (size_t)node * DD + h * DHD;
    const float* a = attn + h * DHD;
    float s = 0.f;
#pragma unroll 8
    for (int j = 0; j < DHD; ++j) s += f[j] * a[j];
    out[i] = s;
}

// Wr[d,h] = sum_j W_dst[d, h*64+j] * attn_r[h,j]   (256x4 folded projection)
__global__ void build_wr(const float* __restrict__ W_dst, const float* __restrict__ attn_r,
                         float* __restrict__ Wr)
{
    int i = blockIdx.x * blockDim.x + threadIdx.x;   // i = d*H + h
    if (i >= DD * HH) return;
    int h = i & (HH - 1);
    int d = i >> 2;
    const float* w = W_dst + (size_t)d * (HH * DHD) + h * DHD;
    const float* a = attn_r + h * DHD;
    float s = 0.f;
#pragma unroll 8
    for (int j = 0; j < DHD; ++j) s += w[j] * a[j];
    Wr[i] = s;
}

// er[n,h] = word[n,:] @ Wr[:,h]   (Wr is tiny -> cache resident)
__global__ void er_kernel(const float* __restrict__ word, const float* __restrict__ Wr,
                          float* __restrict__ er)
{
    int i = blockIdx.x * blockDim.x + threadIdx.x;   // i = n*H + h
    if (i >= NW * HH) return;
    int h = i & (HH - 1);
    int n = i >> 2;
    const float* w = word + (size_t)n * DD;
    float s = 0.f;
#pragma unroll 8
    for (int d = 0; d < DD; ++d) s += w[d] * Wr[d * HH + h];
    er[i] = s;
}

// Order-preserving float <-> uint mapping so atomicMax(u32) == float max
__device__ __forceinline__ unsigned fenc(float f) {
    unsigned b = __float_as_uint(f);
    return (b & 0x80000000u) ? ~b : (b | 0x80000000u);
}
__device__ __forceinline__ float fdec(unsigned u) {
    return __uint_as_float((u & 0x80000000u) ? (u & 0x7FFFFFFFu) : ~u);
}

__global__ void edge_logits(const int* __restrict__ src, const int* __restrict__ dst,
                            const float* __restrict__ el, const float* __restrict__ er,
                            float* __restrict__ ebuf, unsigned* __restrict__ emax)
{
    int i = blockIdx.x * blockDim.x + threadIdx.x;   // i = e*H + h
    if (i >= NE * HH) return;
    int h = i & (HH - 1);
    int e = i >> 2;
    float v = el[src[e] * HH + h] + er[dst[e] * HH + h];
    v = (v > 0.f) ? v : NEG_SLOPE * v;               // leaky_relu
    ebuf[i] = v;
    atomicMax(&emax[dst[e] * HH + h], fenc(v));
}

__global__ void edge_exp(const int* __restrict__ dst, const unsigned* __restrict__ emax,
                         float* __restrict__ ebuf, float* __restrict__ den)
{
    int i = blockIdx.x * blockDim.x + threadIdx.x;   // i = e*H + h
    if (i >= NE * HH) return;
    int h = i & (HH - 1);
    int e = i >> 2;
    float ex = expf(ebuf[i] - fdec(emax[dst[e] * HH + h]));
    ebuf[i] = ex;                                    // reuse buffer for exp values
    atomicAdd(&den[dst[e] * HH + h], ex);
}

// One block per edge: 256 threads cover the H*DH=256 message features.
__global__ __launch_bounds__(256)
void edge_aggregate(const int* __restrict__ src, const int* __restrict__ dst,
                    const float* __restrict__ feat_src, const float* __restrict__ ebuf,
                    const float* __restrict__ den, float* __restrict__ rst)
{
    int e = blockIdx.x;
    int t = threadIdx.x;
    int h = t >> 6;                                  // head = feature/64
    int s = src[e], d = dst[e];
    float alpha = ebuf[(size_t)e * HH + h] / den[d * HH + h];
    atomicAdd(&rst[(size_t)d * DD + t], feat_src[(size_t)s * DD + t] * alpha);
}

// h = word + elu(rst + gat_bias), written in place over rst
__global__ void elu_residual(const float* __restrict__ word, const float* __restrict__ gat_bias,
                             float* __restrict__ rst)
{
    int i = blockIdx.x * blockDim.x + threadIdx.x;
    if (i >= NW * DD) return;
    float u = rst[i] + gat_bias[i & (DD - 1)];
    u = (u > 0.f) ? u : expm1f(u);                   // elu(alpha=1)
    rst[i] = word[i] + u;
}

// ---------------------------------------------------------------------------
extern "C" void kernel_launch(void* const* d_in, const int* in_sizes, int n_in,
                              void* d_out, int out_size, void* d_ws, size_t ws_size,
                              hipStream_t stream)
{
    const float* word   = (const float*)d_in[0];
    const float* sent   = (const float*)d_in[1];
    const int*   src    = (const int*)  d_in[2];
    const int*   dst    = (const int*)  d_in[3];
    const float* W_src  = (const float*)d_in[4];
    const float* W_dst  = (const float*)d_in[5];
    const float* attn_l = (const float*)d_in[6];
    const float* attn_r = (const float*)d_in[7];
    const float* gbias  = (const float*)d_in[8];
    const float* w1     = (const float*)d_in[9];
    const float* b1     = (const float*)d_in[10];
    const float* w2     = (const float*)d_in[11];
    const float* b2     = (const float*)d_in[12];
    float* out = (float*)d_out;

    // ---- workspace layout (256B aligned slabs) ----
    char* base = (char*)d_ws;
    size_t off = 0;
    auto take = [&](size_t bytes) -> void* {
        void* p = base + off;
        off = (off + bytes + 255) & ~(size_t)255;
        return p;
    };
    float*    feat_src = (float*)   take((size_t)NS * DD * sizeof(float));   // 10.2 MB
    float*    el       = (float*)   take((size_t)NS * HH * sizeof(float));
    float*    er       = (float*)   take((size_t)NW * HH * sizeof(float));
    float*    Wr       = (float*)   take((size_t)DD * HH * sizeof(float));
    float*    ebuf     = (float*)   take((size_t)NE * HH * sizeof(float));   // 12.8 MB
    unsigned* emax     = (unsigned*)take((size_t)NW * HH * sizeof(unsigned));
    float*    den      = (float*)   take((size_t)NW * HH * sizeof(float));
    float*    rst      = (float*)   take((size_t)NW * DD * sizeof(float));   // 102.4 MB (rst, then h)
    float*    hid      = (float*)   take((size_t)NW * DF * sizeof(float));   // 204.8 MB

    // ---- 1. feat_src = sent @ W_src  (WMMA fp32, 32x32 tiles) ----
    gemm_wmma_f32<<<dim3(DD / 32, (NS + 31) / 32), 32, 0, stream>>>(
        sent, W_src, nullptr, feat_src, NS, DD, DD, 0);

    // ---- 2. el / er attention logits ----
    el_kernel<<<(NS * HH + 255) / 256, 256, 0, stream>>>(feat_src, attn_l, el, NS * HH);
    build_wr<<<(DD * HH + 255) / 256, 256, 0, stream>>>(W_dst, attn_r, Wr);
    er_kernel<<<(NW * HH + 255) / 256, 256, 0, stream>>>(word, Wr, er);

    // ---- 3. edge softmax + aggregation ----
    hipMemsetAsync(emax, 0, (size_t)NW * HH * sizeof(unsigned), stream);  // fenc-min
    hipMemsetAsync(den,  0, (size_t)NW * HH * sizeof(float), stream);
    hipMemsetAsync(rst,  0, (size_t)NW * DD * sizeof(float), stream);

    edge_logits<<<(NE * HH + 255) / 256, 256, 0, stream>>>(src, dst, el, er, ebuf, emax);
    edge_exp   <<<(NE * HH + 255) / 256, 256, 0, stream>>>(dst, emax, ebuf, den);
    edge_aggregate<<<NE, 256, 0, stream>>>(src, dst, feat_src, ebuf, den, rst);

    // ---- 4. h = word + elu(rst + bias)  (in place over rst) ----
    elu_residual<<<(NW * DD + 255) / 256, 256, 0, stream>>>(word, gbias, rst);

    // ---- 5. FFN: out = relu(h@w1 + b1) @ w2 + b2  (WMMA fp32, fused epilogues) ----
    gemm_wmma_f32<<<dim3(DF / 32, (NW + 31) / 32), 32, 0, stream>>>(
        rst, w1, b1, hid, NW, DF, DD, 1);
    gemm_wmma_f32<<<dim3(DD / 32, (NW + 31) / 32), 32, 0, stream>>>(
        hid, w2, b2, out, NW, DD, DF, 0);
}